// VectorQuantizer_57595511439938
// MI455X (gfx1250) — compile-verified
//
#include <hip/hip_runtime.h>
#include <hip/hip_bf16.h>
#include <math.h>

typedef float v2f __attribute__((ext_vector_type(2)));
typedef float v8f __attribute__((ext_vector_type(8)));

#define VQ_N  65536   // 16*8*512 rows
#define VQ_D  64
#define VQ_K  512

// ---------------------------------------------------------------------------
// Kernel 0: init workspace — zero loss accum + histogram, compute ||w||^2.
// ---------------------------------------------------------------------------
__global__ __launch_bounds__(512) void vq_init_kernel(const float* __restrict__ w,
                                                      float* __restrict__ w2,
                                                      float* __restrict__ sumsq,
                                                      unsigned int* __restrict__ hist) {
    const int t = threadIdx.x;  // 512 threads, one per code
    float s = 0.0f;
    #pragma unroll 8
    for (int d = 0; d < VQ_D; ++d) {
        float v = w[t * VQ_D + d];
        s += v * v;
    }
    w2[t] = s;
    hist[t] = 0u;
    if (t == 0) *sumsq = 0.0f;
}

// ---------------------------------------------------------------------------
// Kernel 1: fused distance + argmin using V_WMMA_F32_16X16X4_F32.
// Each wave (32 lanes) owns a 16-row tile; block = 8 waves = 128 rows.
// A-frag layout (32-bit A, 16x4): lanes 0-15 -> M=lane, K={4kb+0,4kb+1};
//                                 lanes 16-31 -> M=lane-16, K={4kb+2,4kb+3}.
// B-frag layout (32-bit B, 4x16): lanes 0-15 -> N=lane, K={0,1}; lanes 16-31
//                                 -> N=lane-16, K={2,3}  (K-major in lane).
// C layout: VGPR v, lane l -> row = v + 8*(l>=16), col = l & 15.
// ---------------------------------------------------------------------------
__global__ __launch_bounds__(256) void vq_argmin_kernel(const float* __restrict__ x,
                                                        const float* __restrict__ w,
                                                        const float* __restrict__ w2,
                                                        int* __restrict__ idx_out) {
    const int lane = threadIdx.x & 31;
    const int wv   = threadIdx.x >> 5;
    const int rowBase = blockIdx.x * 128 + wv * 16;
    const int half = lane >> 4;   // 0 or 1
    const int l15  = lane & 15;

    // Load all A fragments for this 16-row tile (D=64 -> 16 steps of K=4).
    v2f a[16];
    const float* xrow = x + (size_t)(rowBase + l15) * VQ_D + 2 * half;
    #pragma unroll
    for (int kb = 0; kb < 16; ++kb)
        a[kb] = *(const v2f*)(xrow + kb * 4);

    float bestv[8];
    int   besti[8];
    #pragma unroll
    for (int v = 0; v < 8; ++v) { bestv[v] = 3.0e38f; besti[v] = 0; }

    // Sweep the 512 codes in 32 tiles of 16 columns.
    for (int t = 0; t < 32; ++t) {
        const int col = t * 16 + l15;                 // this lane's column
        const float* wrow = w + (size_t)col * VQ_D + 2 * half;
        v8f c = {0.f, 0.f, 0.f, 0.f, 0.f, 0.f, 0.f, 0.f};
        #pragma unroll
        for (int kb = 0; kb < 16; ++kb) {
            v2f b = *(const v2f*)(wrow + kb * 4);
            // (neg_a, A, neg_b, B, c_mod, C, reuse_a, reuse_b)
            c = __builtin_amdgcn_wmma_f32_16x16x4_f32(
                    false, a[kb], false, b, (short)0, c, false, false);
        }
        const float w2c = w2[col];
        #pragma unroll
        for (int v = 0; v < 8; ++v) {
            float dist = w2c - 2.0f * c[v];           // ||x||^2 dropped (row-const)
            bool lt = dist < bestv[v];
            besti[v] = lt ? col  : besti[v];          // branch-free (cndmask)
            bestv[v] = lt ? dist : bestv[v];
        }
    }

    // Reduce the per-lane minima across the 16 column lanes of each half.
    #pragma unroll
    for (int v = 0; v < 8; ++v) {
        float bv = bestv[v];
        int   bi = besti[v];
        #pragma unroll
        for (int m = 8; m >= 1; m >>= 1) {
            float ov = __shfl_xor(bv, m, 32);
            int   oi = __shfl_xor(bi, m, 32);
            bool take = (ov < bv) || (ov == bv && oi < bi);  // argmin = first min
            bv = take ? ov : bv;
            bi = take ? oi : bi;
        }
        if (l15 == 0)
            idx_out[rowBase + v + 8 * half] = bi;
    }
}

// ---------------------------------------------------------------------------
// Kernel 2: gather codebook rows, write quantized_st + one-hot encodings,
// accumulate sum((q-x)^2) and histogram. Block = 256 threads, 16 rows/block.
// ---------------------------------------------------------------------------
__global__ __launch_bounds__(256) void vq_gather_kernel(const float* __restrict__ x,
                                                        const float* __restrict__ w,
                                                        const int* __restrict__ idx,
                                                        float* __restrict__ qout,
                                                        float* __restrict__ enc,
                                                        float* __restrict__ sumsq,
                                                        unsigned int* __restrict__ hist) {
    __shared__ float red[256];
    const int t = threadIdx.x;
    const int rowBase = blockIdx.x * 16;
    float acc = 0.0f;
    for (int r = 0; r < 16; ++r) {
        const int row = rowBase + r;
        const int ci  = idx[row];
        // one-hot encodings row (512 floats, two coalesced 256-wide stores)
        enc[(size_t)row * VQ_K + t]       = (t == ci)       ? 1.0f : 0.0f;
        enc[(size_t)row * VQ_K + 256 + t] = (t + 256 == ci) ? 1.0f : 0.0f;
        if (t < VQ_D) {
            float q  = w[(size_t)ci * VQ_D + t];
            float xv = x[(size_t)row * VQ_D + t];
            qout[(size_t)row * VQ_D + t] = q;       // quantized_st == quantized
            float d = q - xv;
            acc += d * d;
        }
        if (t == 0) atomicAdd(&hist[ci], 1u);
    }
    red[t] = acc;
    __syncthreads();
    #pragma unroll
    for (int s = 128; s > 0; s >>= 1) {
        if (t < s) red[t] += red[t + s];
        __syncthreads();
    }
    if (t == 0) atomicAdd(sumsq, red[0]);
}

// ---------------------------------------------------------------------------
// Kernel 3: finalize loss and perplexity.
// ---------------------------------------------------------------------------
__global__ __launch_bounds__(512) void vq_finalize_kernel(const float* __restrict__ sumsq,
                                                          const unsigned int* __restrict__ hist,
                                                          float* __restrict__ loss_out,
                                                          float* __restrict__ perp_out) {
    __shared__ float red[512];
    const int t = threadIdx.x;  // 512 threads, one per code
    const float p = (float)hist[t] * (1.0f / (float)VQ_N);
    red[t] = p * logf(p + 1.1920929e-7f);   // FLT_EPSILON, matches jnp eps
    __syncthreads();
    #pragma unroll
    for (int s = 256; s > 0; s >>= 1) {
        if (t < s) red[t] += red[t + s];
        __syncthreads();
    }
    if (t == 0) {
        *perp_out = expf(-red[0]);
        // q_latent + 0.25*e_latent, numerically identical diffs
        *loss_out = 1.25f * (*sumsq) * (1.0f / ((float)VQ_N * (float)VQ_D));
    }
}

// ---------------------------------------------------------------------------
// Host-side launcher.
// d_out layout (flat, return order): [loss | quantized_st (N*D) | perplexity |
//                                     encodings (N*K)]
// ---------------------------------------------------------------------------
extern "C" void kernel_launch(void* const* d_in, const int* in_sizes, int n_in,
                              void* d_out, int out_size, void* d_ws, size_t ws_size,
                              hipStream_t stream) {
    const float* x = (const float*)d_in[0];   // [16,8,512,64] fp32
    const float* w = (const float*)d_in[1];   // [512,64] fp32

    float* out      = (float*)d_out;
    float* loss_out = out;                          // 1
    float* q_out    = out + 1;                      // N*D
    float* perp_out = out + 1 + (size_t)VQ_N * VQ_D;// 1
    float* enc_out  = out + 2 + (size_t)VQ_N * VQ_D;// N*K

    char* ws = (char*)d_ws;
    float*        sumsq = (float*)(ws + 0);         // 4 B
    unsigned int* hist  = (unsigned int*)(ws + 256);// 2 KB
    float*        w2    = (float*)(ws + 4096);      // 2 KB
    int*          idx   = (int*)(ws + 8192);        // 256 KB

    vq_init_kernel   <<<1,            512, 0, stream>>>(w, w2, sumsq, hist);
    vq_argmin_kernel <<<VQ_N / 128,   256, 0, stream>>>(x, w, w2, idx);
    vq_gather_kernel <<<VQ_N / 16,    256, 0, stream>>>(x, w, idx, q_out, enc_out,
                                                        sumsq, hist);
    vq_finalize_kernel<<<1,           512, 0, stream>>>(sumsq, hist, loss_out, perp_out);
}